// TemporalSelfAttention_2860448219659
// MI455X (gfx1250) — compile-verified
//
#include <hip/hip_runtime.h>
#include <hip/hip_bf16.h>

// ---------------------------------------------------------------------------
// TemporalSelfAttention (BEVFormer-style MSDeformAttn, single level, N=1)
//   Lq = Lin = 200*200 = 40000, D_MODEL=256, N_HEADS=8, N_POINTS=4, D_HEAD=32
// Pipeline:
//   [K1] value   = input_flatten @ W_value^T + b_value           (WMMA f32 GEMM)
//   [K2] offattn = (query+query_pos) @ [W_off;W_attn]^T + bias   (WMMA f32 GEMM, N=96)
//   [K3] sampled = softmax + bilinear deformable sampling        (coalesced gathers)
//   [K4] out     = sampled @ W_out^T + b_out                     (WMMA f32 GEMM)
//
// GEMM scheme: block = 8 waves. The block's 64-column (or 32-column) weight
// group is staged once into LDS with coalesced b128 loads; each wave then
// computes a 16-row x GW-column strip with V_WMMA_F32_16X16X4_F32, reading B
// from LDS (padded stride 260 floats -> conflict-free ds_load_b64: bank =
// 4*idx + 2*half covers all 64 banks exactly once per fetch).
// ---------------------------------------------------------------------------

#define H_SP 200
#define W_SP 200
#define LQ (H_SP * W_SP)
#define D_MODEL 256
#define N_HEADS 8
#define N_POINTS 4
#define D_HEAD 32
#define K_DIM 256
#define LDS_STRIDE 260  // 256 + 4 pad floats

typedef __attribute__((ext_vector_type(2))) float v2f;
typedef __attribute__((ext_vector_type(8))) float v8f;

// ---------------------------------------------------------------------------
// C[M x N] = (A0 (+A1 if HAS_A1)) [M x 256] @ W^T + bias
// W row-major [N x 256] (torch Linear layout). Columns [0,n0cols) from
// (W0,bias0), columns [n0cols,N) from (W1,bias1). GW is the per-block column
// group width (must not straddle the n0cols boundary; 64 | n0cols etc).
// grid.x = n_groups * blocks_per_group ; block = 256 threads = 8 waves.
// ---------------------------------------------------------------------------
template <int GW, bool HAS_A1>
__global__ __launch_bounds__(256) void wmma_gemm_f32(
    const float* __restrict__ A0, const float* __restrict__ A1,
    const float* __restrict__ W0, const float* __restrict__ bias0, int n0cols,
    const float* __restrict__ W1, const float* __restrict__ bias1,
    float* __restrict__ C, int N, int tiles_m, int blocks_per_group)
{
    constexpr int NT = GW / 16;              // 16-col tiles per wave
    __shared__ float lds_b[GW * LDS_STRIDE]; // GW=64 -> 65 KB of the 320 KB WGP LDS

    const int group = blockIdx.x / blocks_per_group;
    const int bm    = blockIdx.x - group * blocks_per_group;
    const int gcol0 = group * GW;

    // ---- cooperative, fully-coalesced copy: GW x 256 weight block -> LDS ----
    // The group lies entirely inside one weight matrix, so its rows form one
    // contiguous GW*256-float region.
    const float* __restrict__ wsrc = (gcol0 < n0cols)
        ? (W0 + (size_t)gcol0 * K_DIM)
        : (W1 + (size_t)(gcol0 - n0cols) * K_DIM);
    const float4* __restrict__ wsrc4 = (const float4*)wsrc;
    {
        const int tid = threadIdx.x;
        constexpr int TOTAL4 = GW * K_DIM / 4;       // float4 elements
#pragma unroll
        for (int i = 0; i < TOTAL4 / 256; ++i) {
            const int f  = tid + i * 256;
            const float4 v = wsrc4[f];               // global_load_b128
            const int r  = f >> 6;                   // 64 float4 per 256-float row
            const int k4 = f & 63;
            *(float4*)&lds_b[r * LDS_STRIDE + k4 * 4] = v;  // 16B-aligned (1040*r+16*k4)
        }
    }
    __syncthreads();

    const int wave = threadIdx.x >> 5;
    const int lane = threadIdx.x & 31;
    const int tm   = bm * 8 + wave;
    if (tm >= tiles_m) return;                 // whole-wave exit: EXEC stays all-ones

    const int half = lane >> 4;                // 0: lanes 0-15, 1: lanes 16-31
    const int idx  = lane & 15;
    const int m0   = tm << 4;
    const int row  = m0 + idx;

    const float2* __restrict__ arow0 = (const float2*)(A0 + (size_t)row * K_DIM);
    const float2* __restrict__ arow1 = HAS_A1
        ? (const float2*)(A1 + (size_t)row * K_DIM) : (const float2*)0;
    const float* __restrict__ bbase = &lds_b[idx * LDS_STRIDE + 2 * half];

    v8f acc[NT] = {};
    // K = 256 in steps of 4. f32 WMMA lane layout:
    //   a[j] = A[row][4kk + 2*half + j]   (A 16x4 -> 2 VGPRs)
    //   b[j] = W[col][4kk + 2*half + j]   (B 4x16 -> 2 VGPRs), col = gcol0+16t+idx
#pragma unroll 4
    for (int kk = 0; kk < K_DIM / 4; ++kk) {
        float2 av = arow0[kk * 2 + half];
        if constexpr (HAS_A1) {
            const float2 a1 = arow1[kk * 2 + half];
            av.x += a1.x;
            av.y += a1.y;
        }
        v2f a; a[0] = av.x; a[1] = av.y;
#pragma unroll
        for (int t = 0; t < NT; ++t) {
            const float* bp = bbase + (t * 16) * LDS_STRIDE + kk * 4;  // ds_load_b64
            v2f b; b[0] = bp[0]; b[1] = bp[1];
            acc[t] = __builtin_amdgcn_wmma_f32_16x16x4_f32(
                /*neg_a=*/false, a, /*neg_b=*/false, b,
                /*c_mod=*/(short)0, acc[t], /*reuse_a=*/false, /*reuse_b=*/false);
        }
    }

    // D layout: VGPR j holds row (m0 + j + 8*half), column (gcol0 + 16t + idx)
#pragma unroll
    for (int t = 0; t < NT; ++t) {
        const int col  = gcol0 + t * 16 + idx;
        const float bv = (col < n0cols) ? bias0[col] : bias1[col - n0cols];
#pragma unroll
        for (int j = 0; j < 8; ++j) {
            const int r = m0 + j + 8 * half;
            C[(size_t)r * N + col] = acc[t][j] + bv;
        }
    }
}

// ---------------------------------------------------------------------------
// Deformable sampling + per-(q,h) softmax over the 4 points.
// One wave per (query, head): lane d handles channel d of the 32-dim head.
// All control math is wave-uniform, so corner-validity branches are uniform
// and the value gathers are 128B-coalesced per corner.
// ---------------------------------------------------------------------------
__global__ __launch_bounds__(256) void deform_sample(
    const float* __restrict__ value,    // (LQ, 256)  head h -> cols [32h, 32h+32)
    const float* __restrict__ offattn,  // (LQ, 96)   [0,64) offsets, [64,96) logits
    const float* __restrict__ refpts,   // (LQ, 2)
    float* __restrict__ sampled)        // (LQ, 256)
{
    const int q = blockIdx.x;
    const int h = threadIdx.x >> 5;     // 8 waves == 8 heads
    const int lane = threadIdx.x & 31;  // channel within head

    const float* __restrict__ oa = offattn + (size_t)q * 96;
    const float rx = refpts[q * 2 + 0];
    const float ry = refpts[q * 2 + 1];

    // softmax over the 4 point logits
    const float l0 = oa[64 + h * 4 + 0];
    const float l1 = oa[64 + h * 4 + 1];
    const float l2 = oa[64 + h * 4 + 2];
    const float l3 = oa[64 + h * 4 + 3];
    const float mx = fmaxf(fmaxf(l0, l1), fmaxf(l2, l3));
    const float e0 = __expf(l0 - mx);
    const float e1 = __expf(l1 - mx);
    const float e2 = __expf(l2 - mx);
    const float e3 = __expf(l3 - mx);
    const float inv = 1.0f / (e0 + e1 + e2 + e3);
    const float aw[N_POINTS] = { e0 * inv, e1 * inv, e2 * inv, e3 * inv };

    const float bx = rx * (float)W_SP - 0.5f;  // loc.x*W - 0.5 = rx*W + ox - 0.5
    const float by = ry * (float)H_SP - 0.5f;

    float acc = 0.0f;
#pragma unroll
    for (int p = 0; p < N_POINTS; ++p) {
        const float x = bx + oa[h * 8 + 2 * p + 0];
        const float y = by + oa[h * 8 + 2 * p + 1];
        const float x0f = floorf(x);
        const float y0f = floorf(y);
        const int x0 = (int)x0f;
        const int y0 = (int)y0f;
        const float lw = x - x0f;
        const float lh = y - y0f;

        const float cw[4] = { (1.0f - lh) * (1.0f - lw), (1.0f - lh) * lw,
                              lh * (1.0f - lw),          lh * lw };
        const int cx[4] = { x0, x0 + 1, x0,     x0 + 1 };
        const int cy[4] = { y0, y0,     y0 + 1, y0 + 1 };

#pragma unroll
        for (int c = 0; c < 4; ++c) {
            const int xi = cx[c], yi = cy[c];
            if (xi >= 0 && xi < W_SP && yi >= 0 && yi < H_SP) {  // wave-uniform
                const size_t pix = (size_t)(yi * W_SP + xi);
                const float g = value[pix * D_MODEL + h * D_HEAD + lane];
                acc += aw[p] * cw[c] * g;
            }
        }
    }
    sampled[(size_t)q * D_MODEL + h * D_HEAD + lane] = acc;
}

// ---------------------------------------------------------------------------
// Host-side launcher. Input order (setup_inputs):
//  0 query  1 query_pos  2 reference_points  3 input_flatten
//  4 W_value 5 b_value  6 W_off 7 b_off  8 W_attn 9 b_attn  10 W_out 11 b_out
// ---------------------------------------------------------------------------
extern "C" void kernel_launch(void* const* d_in, const int* in_sizes, int n_in,
                              void* d_out, int out_size, void* d_ws, size_t ws_size,
                              hipStream_t stream) {
    const float* query     = (const float*)d_in[0];
    const float* query_pos = (const float*)d_in[1];
    const float* refpts    = (const float*)d_in[2];
    const float* inflat    = (const float*)d_in[3];
    const float* W_value   = (const float*)d_in[4];
    const float* b_value   = (const float*)d_in[5];
    const float* W_off     = (const float*)d_in[6];
    const float* b_off     = (const float*)d_in[7];
    const float* W_attn    = (const float*)d_in[8];
    const float* b_attn    = (const float*)d_in[9];
    const float* W_out     = (const float*)d_in[10];
    const float* b_out     = (const float*)d_in[11];
    float* out = (float*)d_out;

    float* ws = (float*)d_ws;
    float* value   = ws;                              // LQ*256 floats
    float* offattn = value + (size_t)LQ * D_MODEL;    // LQ*96  floats
    float* sampled = offattn + (size_t)LQ * 96;       // LQ*256 floats

    const int tiles_m = LQ / 16;                      // 2500
    const int bpg = (tiles_m + 7) / 8;                // 313 blocks per column group

    // [K1] value projection: 40000 x 256 x 256  (4 groups of 64 cols)
    wmma_gemm_f32<64, false><<<4 * bpg, 256, 0, stream>>>(
        inflat, nullptr, W_value, b_value, D_MODEL, W_value, b_value,
        value, D_MODEL, tiles_m, bpg);

    // [K2] fused offsets(64) + attn logits(32): 40000 x 96 x 256, A = q + qpos
    // (3 groups of 32 cols; the 64-col W_off/W_attn boundary is group-aligned)
    wmma_gemm_f32<32, true><<<3 * bpg, 256, 0, stream>>>(
        query, query_pos, W_off, b_off, 64, W_attn, b_attn,
        offattn, 96, tiles_m, bpg);

    // [K3] softmax + bilinear deformable sampling
    deform_sample<<<LQ, 256, 0, stream>>>(value, offattn, refpts, sampled);

    // [K4] output projection: 40000 x 256 x 256
    wmma_gemm_f32<64, false><<<4 * bpg, 256, 0, stream>>>(
        sampled, nullptr, W_out, b_out, D_MODEL, W_out, b_out,
        out, D_MODEL, tiles_m, bpg);
}